// Patcher_19258633356160
// MI455X (gfx1250) — compile-verified
//
#include <hip/hip_runtime.h>

// Problem constants (match reference)
#define BB   16
#define NN   16
#define HH   896
#define WW   896
#define CCH  3
#define PHH  300
#define PWW  300
#define PPAD 304          // padded weight-vector length (multiple of 16, covers 19 tiles)
#define MIN_PH_F 60.0f

// d_out layout (flat concat in return order, all float32):
//   imgs_out [B,H,W,C]          offset 0
//   pboxes   [B,N,4]            offset O_PBOX
//   valid    [B,N]              offset O_VALID
//   td       [B,N,3]            offset O_TD
//   grads    [PH,PW,C]          offset O_GRADS
#define O_PBOX  (BB*HH*WW*CCH)
#define O_VALID (O_PBOX + BB*NN*4)
#define O_TD    (O_VALID + BB*NN)
#define O_GRADS (O_TD + BB*NN*3)

// d_ws layout:
//   int params (SoA):  yp[256] | xp[256] | ph[256] | pw[256] | valid[256]   (5120 B)
//   at byte 8192:      Ay[256][PPAD] floats
//   then:              Ax[256][PPAD] floats
#define WS_WEIGHTS_OFF 8192

typedef __attribute__((ext_vector_type(2))) float v2f;
typedef __attribute__((ext_vector_type(8))) float v8f;

// ---------------- Kernel 1: boxes -> pboxes / valid / td copy / int windows ----------
__global__ void k_boxes(const float* __restrict__ boxes,
                        const int* __restrict__ td,
                        float* __restrict__ out,
                        int* __restrict__ wsi) {
    int tid = threadIdx.x;
    if (tid < BB * NN) {
        float ymin = boxes[tid*4+0], xmin = boxes[tid*4+1];
        float ymax = boxes[tid*4+2], xmax = boxes[tid*4+3];
        float h = ymax - ymin, w = xmax - xmin;
        float pw = h * 0.5f;          // SCALE = 0.5
        float ph = pw;                // ASPECT = 1.0
        float oy = ymin + h * 0.5f, ox = xmin + w * 0.5f;
        float yp = fmaxf(oy - ph * 0.5f, 0.0f);
        float xp = fmaxf(ox - pw * 0.5f, 0.0f);
        yp = (yp + ph > (float)HH) ? ((float)HH - ph) : yp;
        xp = (xp + pw > (float)WW) ? ((float)WW - pw) : xp;
        out[O_PBOX + tid*4+0] = yp;
        out[O_PBOX + tid*4+1] = xp;
        out[O_PBOX + tid*4+2] = ph;
        out[O_PBOX + tid*4+3] = pw;
        int valid = (ph > MIN_PH_F) ? 1 : 0;
        out[O_VALID + tid] = valid ? 1.0f : 0.0f;
        int ypi = (int)yp, xpi = (int)xp;          // trunc toward zero (values >= 0)
        int phi = (int)ph; if (phi < 1) phi = 1;
        int pwi = (int)pw; if (pwi < 1) pwi = 1;
        wsi[       tid] = ypi;
        wsi[ 256 + tid] = xpi;
        wsi[ 512 + tid] = phi;
        wsi[ 768 + tid] = pwi;
        wsi[1024 + tid] = valid;
    }
    if (tid < BB * NN * 3) {
        out[O_TD + tid] = td[tid] ? 1.0f : 0.0f;
    }
}

// Sum of bilinear weights hitting input index p from all output rows in the window.
// u(r) = (r + 0.5 - off) * (full/len) - 0.5 ; weight = max(0, 1 - |u - p|).
__device__ __forceinline__ float axis_weight_sum(int p, int off, int len, int full) {
    float s   = (float)len / (float)full;      // scale < 1 (downsampling)
    float inv = (float)full / (float)len;
    float rc  = (float)off + ((float)p + 0.5f) * s - 0.5f;  // r where u == p
    int r0 = (int)floorf(rc) - 2;
    float acc = 0.0f;
    #pragma unroll
    for (int i = 0; i < 5; ++i) {              // |u-p|<1 => r in (rc-s, rc+s), s<1: covered
        int r = r0 + i;
        if (r >= off && r < off + len) {
            float u = ((float)r + 0.5f - (float)off) * inv - 0.5f;
            acc += fmaxf(1.0f - fabsf(u - (float)p), 0.0f);
        }
    }
    return acc;
}

// ---------------- Kernel 2: per-box separable VJP weight vectors ----------
__global__ void k_weights(const int* __restrict__ wsi,
                          float* __restrict__ Ay,
                          float* __restrict__ Ax) {
    int box = blockIdx.x;
    int p = threadIdx.x;
    if (p >= PPAD) return;
    int ypi = wsi[box], xpi = wsi[256+box];
    int phi = wsi[512+box], pwi = wsi[768+box];
    int val = wsi[1024+box];
    float ay = 0.0f, ax = 0.0f;
    if (val && p < PHH) {
        ay = axis_weight_sum(p, ypi, phi, PHH);
        ax = axis_weight_sum(p, xpi, pwi, PWW);
    }
    Ay[box*PPAD + p] = ay;   // zero pad + zero for invalid boxes
    Ax[box*PPAD + p] = ax;
}

// ---------------- Kernel 3: image placement (last valid covering box wins) ----------
__global__ void k_place(const float* __restrict__ images,
                        const float* __restrict__ patch,
                        const int* __restrict__ wsi,
                        float* __restrict__ out) {
    int tid = blockIdx.x * blockDim.x + threadIdx.x;   // B*H*W = 12,845,056 < 2^31
    const int total = BB * HH * WW;
    if (tid >= total) return;
    int b   = tid / (HH * WW);
    int rem = tid % (HH * WW);
    int y = rem / WW, x = rem % WW;

    int found = -1;
    #pragma unroll
    for (int n = NN - 1; n >= 0; --n) {                // sequential scan: last write wins
        int idx = b * NN + n;
        if (found < 0 && wsi[1024 + idx]) {
            int ypi = wsi[idx], xpi = wsi[256+idx];
            int phi = wsi[512+idx], pwi = wsi[768+idx];
            if (y >= ypi && y < ypi + phi && x >= xpi && x < xpi + pwi) found = idx;
        }
    }
    int o = tid * 3;
    if (found >= 0) {
        int ypi = wsi[found], xpi = wsi[256+found];
        int phi = wsi[512+found], pwi = wsi[768+found];
        float u = ((float)y + 0.5f - (float)ypi) * ((float)PHH / (float)phi) - 0.5f;
        float v = ((float)x + 0.5f - (float)xpi) * ((float)PWW / (float)pwi) - 0.5f;
        int p0 = (int)floorf(u); p0 = min(max(p0, 0), PHH - 2);
        int q0 = (int)floorf(v); q0 = min(max(q0, 0), PWW - 2);
        float fy = fminf(fmaxf(u - (float)p0, 0.0f), 1.0f);
        float fx = fminf(fmaxf(v - (float)q0, 0.0f), 1.0f);
        const float* r0p = patch + (p0 * PWW + q0) * 3;
        const float* r1p = r0p + PWW * 3;
        #pragma unroll
        for (int c = 0; c < 3; ++c) {
            float v00 = r0p[c], v01 = r0p[3 + c];
            float v10 = r1p[c], v11 = r1p[3 + c];
            float top = v00 + (v01 - v00) * fx;
            float bot = v10 + (v11 - v10) * fx;
            out[o + c] = top + (bot - top) * fy;
        }
    } else {
        out[o + 0] = images[o + 0];
        out[o + 1] = images[o + 1];
        out[o + 2] = images[o + 2];
    }
}

// ---------------- Kernel 4: grads via V_WMMA_F32_16X16X4_F32 ----------
// grads[p,q,c] = sum_b ( sum_n Ay[b,n,p]*Ax[b,n,q] ) * hgrad[b,p,q,c]
// Per (b): S = A(300x16) x B(16x300), A[p][n]=Ay[b,n,p], B[n][q]=Ax[b,n,q].
// One wave per 16x16 output tile; K=16 done as 4 WMMA steps of K=4.
__global__ void __launch_bounds__(32) k_grads(const float* __restrict__ Ay,
                                              const float* __restrict__ Ax,
                                              const float* __restrict__ hgrad,
                                              float* __restrict__ out) {
    int p0 = blockIdx.x * 16;
    int q0 = blockIdx.y * 16;
    int lane = threadIdx.x;        // wave32
    int half = lane >> 4;          // 0: K even pair, 1: K odd pair (per ISA 32-bit A layout)
    int lm   = lane & 15;

    float acc[8][3];
    #pragma unroll
    for (int r = 0; r < 8; ++r) { acc[r][0] = 0.f; acc[r][1] = 0.f; acc[r][2] = 0.f; }

    for (int b = 0; b < BB; ++b) {
        const float* Ayb = Ay + (b * NN) * PPAD;
        const float* Axb = Ax + (b * NN) * PPAD;
        v8f c = {};
        #pragma unroll
        for (int kk = 0; kk < NN; kk += 4) {
            int k0 = kk + 2 * half;
            v2f a, bv;
            // A 16x4 f32: lanes0-15 hold (M=lm, K=kk+0/kk+1), lanes16-31 (M=lm, K=kk+2/kk+3)
            a.x  = Ayb[(k0 + 0) * PPAD + p0 + lm];
            a.y  = Ayb[(k0 + 1) * PPAD + p0 + lm];
            // B 4x16 f32: lanes0-15 hold (K=kk+0/kk+1, N=lm), lanes16-31 (K=kk+2/kk+3, N=lm)
            bv.x = Axb[(k0 + 0) * PPAD + q0 + lm];
            bv.y = Axb[(k0 + 1) * PPAD + q0 + lm];
            c = __builtin_amdgcn_wmma_f32_16x16x4_f32(false, a, false, bv,
                                                      (short)0, c, false, false);
        }
        // D layout: vgpr r -> (M = r + 8*half, N = lm). Branch-free masking keeps EXEC full.
        int q  = q0 + lm;
        int qc = min(q, PWW - 1);
        #pragma unroll
        for (int r = 0; r < 8; ++r) {
            int p  = p0 + r + 8 * half;
            int pc = min(p, PHH - 1);
            float m = (p < PHH && q < PWW) ? 1.0f : 0.0f;
            const float* hg = hgrad + (((b * PHH + pc) * PWW) + qc) * 3;
            float s = c[r] * m;
            acc[r][0] += s * hg[0];
            acc[r][1] += s * hg[1];
            acc[r][2] += s * hg[2];
        }
    }
    int q = q0 + lm;
    if (q < PWW) {
        #pragma unroll
        for (int r = 0; r < 8; ++r) {
            int p = p0 + r + 8 * half;
            if (p < PHH) {
                float* og = out + O_GRADS + (p * PWW + q) * 3;
                og[0] = acc[r][0]; og[1] = acc[r][1]; og[2] = acc[r][2];
            }
        }
    }
}

extern "C" void kernel_launch(void* const* d_in, const int* in_sizes, int n_in,
                              void* d_out, int out_size, void* d_ws, size_t ws_size,
                              hipStream_t stream) {
    const float* boxes   = (const float*)d_in[0];
    const float* images  = (const float*)d_in[1];
    const float* patch   = (const float*)d_in[2];
    const float* hgrad   = (const float*)d_in[3];
    const int*   td      = (const int*)d_in[4];
    float* out = (float*)d_out;

    int*   wsi = (int*)d_ws;
    float* Ay  = (float*)((char*)d_ws + WS_WEIGHTS_OFF);
    float* Ax  = Ay + 256 * PPAD;

    // K1: box math + td passthrough (1 block, 1024 threads)
    k_boxes<<<1, 1024, 0, stream>>>(boxes, td, out, wsi);

    // K2: separable weight vectors, one block per box
    k_weights<<<BB * NN, 320, 0, stream>>>(wsi, Ay, Ax);

    // K3: image placement, one thread per pixel (divides exactly: 12845056/256)
    const int total = BB * HH * WW;
    k_place<<<total / 256, 256, 0, stream>>>(images, patch, wsi, out);

    // K4: WMMA grads, one wave per 16x16 tile of the 300x300 patch
    dim3 gg((PHH + 15) / 16, (PWW + 15) / 16);
    k_grads<<<gg, 32, 0, stream>>>(Ay, Ax, hgrad, out);
}